// DepthwiseShortConv1D_27247272525827
// MI455X (gfx1250) — compile-verified
//
#include <hip/hip_runtime.h>
#include <cstdint>
#include <cstddef>

// ---------------------------------------------------------------------------
// DepthwiseShortConv1D + SiLU, MI455X (gfx1250)
//   y[b,t,c] = silu( sum_{j=0..3} kernel[j,c] * x[b,t-j,c] )   (zero history)
//   next_cache[b,j,c] = x[b, T-3+j, c]
// Memory-bound (AI ~1.4 flop/B, floor ~11.5us @ 23.3 TB/s). Strategy:
// float4-per-thread channel columns (perfect b128 coalescing), register
// sliding window along T (1 load + 1 store per element = roofline minimum),
// async global->LDS double-buffered staging (CDNA5 data-mover path).
// WMMA is deliberately absent: depthwise conv is a *diagonal* contraction
// (no channel mixing), which a 16x16 WMMA cannot express without 16x
// redundant work — and the kernel is bandwidth-bound anyway.
// ---------------------------------------------------------------------------

namespace {
constexpr int B_ = 4;
constexpr int T_ = 4096;
constexpr int C_ = 2048;
constexpr int NCG = C_ / 4;       // 512 float4 channel columns
constexpr int BLK = 256;          // threads per block (8 waves)
constexpr int TCHUNK = 128;       // timesteps per block
constexpr int TW = 8;             // timesteps per async buffer
constexpr int NW = TCHUNK / TW;   // 16 buffer fills
}

#if defined(__gfx1250__) && __has_builtin(__builtin_amdgcn_global_load_async_to_lds_b128)
#define USE_ASYNC_LDS 1
#else
#define USE_ASYNC_LDS 0
#endif

__device__ __forceinline__ float silu_f(float v) {
  float e = __expf(-v);
#if __has_builtin(__builtin_amdgcn_rcpf)
  return v * __builtin_amdgcn_rcpf(1.0f + e);
#else
  return v / (1.0f + e);
#endif
}

#if USE_ASYNC_LDS
// Builtin signature (per hipcc diagnostic): params are pointers to
// 'int __attribute__((vector_size(16)))' in global / LDS address spaces.
typedef int v4i_t __attribute__((vector_size(16)));
typedef __attribute__((address_space(1))) v4i_t as1_v4i;
typedef __attribute__((address_space(3))) v4i_t as3_v4i;

__device__ __forceinline__ void async_fill_b128(const float* src, float4* lds_dst) {
  __builtin_amdgcn_global_load_async_to_lds_b128(
      (as1_v4i*)src,
      (as3_v4i*)lds_dst,
      /*imm offset*/ 0, /*cpol*/ 0);
}
#endif

__global__ __launch_bounds__(BLK)
void dwconv_silu_kernel(const float* __restrict__ x,
                        const float* __restrict__ kern,
                        float* __restrict__ y) {
#if USE_ASYNC_LDS
  __shared__ float4 tile[2][TW][BLK];   // 64 KB, double-buffered
#endif
  const int tid = threadIdx.x;
  const int cg  = blockIdx.x * BLK + tid;   // float4 channel column 0..511
  const int c   = cg << 2;
  const int b   = blockIdx.z;
  const int t0  = blockIdx.y * TCHUNK;

  // Per-channel taps (tiny, L2-resident)
  const float4 k0 = *(const float4*)(kern + 0 * C_ + c);
  const float4 k1 = *(const float4*)(kern + 1 * C_ + c);
  const float4 k2 = *(const float4*)(kern + 2 * C_ + c);
  const float4 k3 = *(const float4*)(kern + 3 * C_ + c);

  const size_t colBase = (size_t)b * T_ * C_ + (size_t)c;  // x[b,0,c]

  // 3-deep history window (zeros at t=0 per reference's zero cache)
  float4 p1, p2, p3;
  if (t0 == 0) {
    p1 = p2 = p3 = make_float4(0.f, 0.f, 0.f, 0.f);
  } else {
    p1 = *(const float4*)(x + colBase + (size_t)(t0 - 1) * C_);
    p2 = *(const float4*)(x + colBase + (size_t)(t0 - 2) * C_);
    p3 = *(const float4*)(x + colBase + (size_t)(t0 - 3) * C_);
  }

#if USE_ASYNC_LDS
  // Prologue: queue buffer 0
#pragma unroll
  for (int tt = 0; tt < TW; ++tt)
    async_fill_b128(x + colBase + (size_t)(t0 + tt) * C_, &tile[0][tt][tid]);

  for (int w = 0; w < NW; ++w) {
    const int buf = w & 1;
    if (w + 1 < NW) {
      const int tn = t0 + (w + 1) * TW;
#pragma unroll
      for (int tt = 0; tt < TW; ++tt)
        async_fill_b128(x + colBase + (size_t)(tn + tt) * C_,
                        &tile[buf ^ 1][tt][tid]);
      // async loads retire in order: <=8 outstanding => buffer `buf` complete
      asm volatile("s_wait_asynccnt 8" ::: "memory");
    } else {
      asm volatile("s_wait_asynccnt 0" ::: "memory");
    }
    const int tb = t0 + w * TW;
#pragma unroll
    for (int tt = 0; tt < TW; ++tt) {
      const float4 cur = tile[buf][tt][tid];   // ds_load_b128 (own data only)
      float4 o;
      o.x = fmaf(k3.x, p3.x, fmaf(k2.x, p2.x, fmaf(k1.x, p1.x, k0.x * cur.x)));
      o.y = fmaf(k3.y, p3.y, fmaf(k2.y, p2.y, fmaf(k1.y, p1.y, k0.y * cur.y)));
      o.z = fmaf(k3.z, p3.z, fmaf(k2.z, p2.z, fmaf(k1.z, p1.z, k0.z * cur.z)));
      o.w = fmaf(k3.w, p3.w, fmaf(k2.w, p2.w, fmaf(k1.w, p1.w, k0.w * cur.w)));
      o.x = silu_f(o.x); o.y = silu_f(o.y); o.z = silu_f(o.z); o.w = silu_f(o.w);
      *(float4*)(y + colBase + (size_t)(tb + tt) * C_) = o;  // global_store_b128
      p3 = p2; p2 = p1; p1 = cur;
    }
  }
#else
  // Fallback: direct b128 streaming with register sliding window
  for (int t = t0; t < t0 + TCHUNK; ++t) {
    const float4 cur = *(const float4*)(x + colBase + (size_t)t * C_);
    float4 o;
    o.x = fmaf(k3.x, p3.x, fmaf(k2.x, p2.x, fmaf(k1.x, p1.x, k0.x * cur.x)));
    o.y = fmaf(k3.y, p3.y, fmaf(k2.y, p2.y, fmaf(k1.y, p1.y, k0.y * cur.y)));
    o.z = fmaf(k3.z, p3.z, fmaf(k2.z, p2.z, fmaf(k1.z, p1.z, k0.z * cur.z)));
    o.w = fmaf(k3.w, p3.w, fmaf(k2.w, p2.w, fmaf(k1.w, p1.w, k0.w * cur.w)));
    o.x = silu_f(o.x); o.y = silu_f(o.y); o.z = silu_f(o.z); o.w = silu_f(o.w);
    *(float4*)(y + colBase + (size_t)t * C_) = o;
    p3 = p2; p2 = p1; p1 = cur;
  }
#endif
}

// next_cache[b, j, c] = x[b, T-3+j, c] ;  b in [0,4), j in [0,3)
__global__ void cache_copy_kernel(const float* __restrict__ x,
                                  float* __restrict__ cache) {
  const int idx = blockIdx.x * blockDim.x + threadIdx.x;  // float4 index
  if (idx >= B_ * 3 * NCG) return;
  const int cg = idx % NCG;
  const int bj = idx / NCG;
  const int j = bj % 3;
  const int b = bj / 3;
  const float4 v = *(const float4*)(x + ((size_t)b * T_ + (size_t)(T_ - 3 + j)) * C_ + cg * 4);
  *(float4*)(cache + (size_t)(b * 3 + j) * C_ + cg * 4) = v;
}

extern "C" void kernel_launch(void* const* d_in, const int* in_sizes, int n_in,
                              void* d_out, int out_size, void* d_ws, size_t ws_size,
                              hipStream_t stream) {
  const float* x = (const float*)d_in[0];
  const float* k = (const float*)d_in[1];
  float* y = (float*)d_out;
  float* cache = y + (size_t)B_ * T_ * C_;

  dim3 grid(NCG / BLK, T_ / TCHUNK, B_);   // (2, 32, 4)
  dwconv_silu_kernel<<<grid, BLK, 0, stream>>>(x, k, y);

  const int nf4 = B_ * 3 * NCG;            // 6144 float4s
  cache_copy_kernel<<<(nf4 + 255) / 256, 256, 0, stream>>>(x, cache);
}